// LinearRNN_7095285973545
// MI455X (gfx1250) — compile-verified
//
#include <hip/hip_runtime.h>

// ---------------- problem constants ----------------
#define TLEN 262144
#define L1   128               // level-1 chunk length
#define G1   (TLEN / L1)       // 2048 chunks
#define L2   64                // level-2 chunk length (over chunk boundary states)
#define G2   (G1 / L2)         // 32 super-chunks

typedef float v2f __attribute__((ext_vector_type(2)));
typedef float v8f __attribute__((ext_vector_type(8)));

// ---------------- WMMA f32 16x16x4 fragment helpers ----------------
// A-operand (MxK row-major) OR B-operand supplied as NxK rows:
// lane&15 selects the row, lane>>4 selects the K pair (K=2*hi, 2*hi+1).
__device__ __forceinline__ v2f frag_mk(const float* __restrict__ p, int ld, int lane) {
  const float* q = p + (lane & 15) * ld + 2 * (lane >> 4);
  v2f r; r.x = q[0]; r.y = q[1]; return r;
}
// B-operand supplied as KxN row-major: lane&15 selects column n,
// lane>>4 selects K pair; v0=B[2hi][n], v1=B[2hi+1][n].
__device__ __forceinline__ v2f frag_kn(const float* __restrict__ p, int ld, int lane) {
  const float* q = p + (2 * (lane >> 4)) * ld + (lane & 15);
  v2f r; r.x = q[0]; r.y = q[ld]; return r;
}
__device__ __forceinline__ v8f wmma4(v2f a, v2f b, v8f c) {
  // D = A(16x4) x B(4x16) + C, fp32 in/out  ->  v_wmma_f32_16x16x4_f32
  return __builtin_amdgcn_wmma_f32_16x16x4_f32(false, a, false, b, (short)0, c,
                                               false, false);
}
// C/D tile layout: vgpr r, lanes 0-15 -> M=r, lanes 16-31 -> M=8+r; N=lane&15.
__device__ __forceinline__ void store_tile(float* __restrict__ out, int ld, v8f acc,
                                           int lane) {
  int ln = lane & 15, hi = lane >> 4;
#pragma unroll
  for (int r = 0; r < 8; ++r) out[(r + 8 * hi) * ld + ln] = acc[r];
}

// SGPR broadcast of lane k's value (k is a compile-time constant in all uses):
// v_readlane_b32 -> s, consumed as scalar operand by v_fmac_f32. Keeps the
// latency-critical scan chain out of the DS pipeline entirely.
__device__ __forceinline__ float bcast(float v, int k) {
  return __int_as_float(__builtin_amdgcn_readlane(__float_as_int(v), k));
}

// ---------------- table init: table[0]=I, table[1]=src ----------------
__global__ void __launch_bounds__(256) k_init_pow(const float* __restrict__ src,
                                                  float* __restrict__ table) {
  int idx = blockIdx.x * blockDim.x + threadIdx.x;  // 0..8191
  if (idx < 4096)
    table[idx] = ((idx >> 6) == (idx & 63)) ? 1.0f : 0.0f;
  else if (idx < 8192)
    table[idx] = src[idx - 4096];
}

// ---------------- 64x64x64 fp32 matmul (one workgroup = 4 waves) --------------
// O[blk] = Lm @ R[rStart+blk]   (all 64x64 row-major, stride 4096 floats)
__global__ void __launch_bounds__(128) k_mm64(const float* __restrict__ Lm,
                                              const float* __restrict__ Rbase,
                                              float* __restrict__ Obase, int rStart) {
  const float* R = Rbase + (size_t)(rStart + blockIdx.x) * 4096;
  float* O = Obase + (size_t)blockIdx.x * 4096;
  int lane = threadIdx.x & 31;
  int w = threadIdx.x >> 5;  // wave -> rows 16w..16w+15
  const float* Arow = Lm + w * 16 * 64;
  v8f acc[4] = {};
#pragma unroll
  for (int k0 = 0; k0 < 64; k0 += 4) {
    v2f a = frag_mk(Arow + k0, 64, lane);
#pragma unroll
    for (int nt = 0; nt < 4; ++nt) {
      v2f b = frag_kn(R + k0 * 64 + nt * 16, 64, lane);
      acc[nt] = wmma4(a, b, acc[nt]);
    }
  }
#pragma unroll
  for (int nt = 0; nt < 4; ++nt) store_tile(O + w * 16 * 64 + nt * 16, 64, acc[nt], lane);
}

// ---------------- K1: Bu = u @ B^T  (one wave = 16 timesteps x 64) ------------
__global__ void __launch_bounds__(256) k_bu(const float* __restrict__ u,
                                            const float* __restrict__ B,
                                            float* __restrict__ Bu) {
  int wave = (blockIdx.x * blockDim.x + threadIdx.x) >> 5;
  int lane = threadIdx.x & 31;
  size_t t0 = (size_t)wave * 16;
  const float* Abase = u + t0 * 64;
  v8f acc[4] = {};
#pragma unroll
  for (int k0 = 0; k0 < 64; k0 += 4) {
    v2f a = frag_mk(Abase + k0, 64, lane);
#pragma unroll
    for (int nt = 0; nt < 4; ++nt) {
      v2f b = frag_mk(B + nt * 16 * 64 + k0, 64, lane);  // B rows as NxK
      acc[nt] = wmma4(a, b, acc[nt]);
    }
  }
#pragma unroll
  for (int nt = 0; nt < 4; ++nt) store_tile(Bu + t0 * 64 + nt * 16, 64, acc[nt], lane);
}

// ---------------- parallel chunked scan: one wave per chunk -------------------
// seq holds drive terms; overwritten in place with zero-init local states
// loc_j (pre-update).  End state (loc_len) -> Eout[chunk].  If init!=0 it seeds
// chunk 0 (used only for the final single-chunk level, folding x0 in).
// Transition matrix rows for lane l and l+32 live in VGPRs; x broadcast via
// v_readlane -> SGPR (constant lane indices, EXEC all ones).
__global__ void __launch_bounds__(256) k_scan(const float* __restrict__ mat,
                                              float* __restrict__ seq,
                                              float* __restrict__ Eout,
                                              const float* __restrict__ init,
                                              int len, int nchunks) {
  int wave = (blockIdx.x * blockDim.x + threadIdx.x) >> 5;
  int lane = threadIdx.x & 31;
  if (wave >= nchunks) return;
  float a0[64], a1[64];
#pragma unroll
  for (int q = 0; q < 16; ++q) {
    float4 r0 = *(const float4*)(mat + lane * 64 + q * 4);
    float4 r1 = *(const float4*)(mat + (lane + 32) * 64 + q * 4);
    a0[q * 4 + 0] = r0.x; a0[q * 4 + 1] = r0.y; a0[q * 4 + 2] = r0.z; a0[q * 4 + 3] = r0.w;
    a1[q * 4 + 0] = r1.x; a1[q * 4 + 1] = r1.y; a1[q * 4 + 2] = r1.z; a1[q * 4 + 3] = r1.w;
  }
  float xa = 0.0f, xb = 0.0f;
  if (init && wave == 0) { xa = init[lane]; xb = init[lane + 32]; }
  float* chunk = seq + (size_t)wave * len * 64;
  for (int j = 0; j < len; ++j) {
    float bua = chunk[j * 64 + lane];        // read drive first,
    float bub = chunk[j * 64 + 32 + lane];
    chunk[j * 64 + lane] = xa;               // then overwrite slot with loc_j
    chunk[j * 64 + 32 + lane] = xb;
    float na = bua, nb = bub;
#pragma unroll
    for (int k = 0; k < 32; ++k) {
      float xk = bcast(xa, k);
      na = fmaf(a0[k], xk, na);
      nb = fmaf(a1[k], xk, nb);
    }
#pragma unroll
    for (int k = 0; k < 32; ++k) {
      float xk = bcast(xb, k);
      na = fmaf(a0[32 + k], xk, na);
      nb = fmaf(a1[32 + k], xk, nb);
    }
    xa = na; xb = nb;
  }
  if (Eout) {
    Eout[(size_t)wave * 64 + lane] = xa;
    Eout[(size_t)wave * 64 + 32 + lane] = xb;
  }
}

// ---------------- reconstruct chunk-initial states ----------------------------
// X0[g] = loc2[g] + Pow2[g%64] @ Z[g/64]     (g = 0..G1-1)
__global__ void __launch_bounds__(64) k_x0(const float* __restrict__ Pow2,
                                           const float* __restrict__ Z,
                                           const float* __restrict__ loc2,
                                           float* __restrict__ X0) {
  int g = blockIdx.x, i = threadIdx.x;
  int s = g >> 6, j = g & 63;
  const float* M = Pow2 + (size_t)j * 4096 + (size_t)i * 64;
  const float* z = Z + (size_t)s * 64;
  float acc = loc2[(size_t)g * 64 + i];
#pragma unroll
  for (int k = 0; k < 64; ++k) acc = fmaf(M[k], z[k], acc);
  X0[(size_t)g * 64 + i] = acc;
}

// ---------------- fused epilogue: ys = Loc C^T + X0 W^T + u D^T ---------------
// Wave -> (g-tile gt of 16 chunks, intra-chunk position j). Output rows
// t = (g0+m)*L1 + j, all 64 output columns (4 WMMA N-tiles). W[j*64+p][k]
// = (C @ A^j)[p][k] injects C(A^j x0_g). d_out written exactly once.
__global__ void __launch_bounds__(256) k_out(const float* __restrict__ u,
                                             const float* __restrict__ Loc,
                                             const float* __restrict__ X0,
                                             const float* __restrict__ W,
                                             const float* __restrict__ C,
                                             const float* __restrict__ D,
                                             float* __restrict__ out) {
  int wave = (blockIdx.x * blockDim.x + threadIdx.x) >> 5;
  int lane = threadIdx.x & 31;
  int gt = wave >> 7;                 // 0..G1/16-1
  int j = wave & (L1 - 1);            // 0..L1-1
  int g0 = gt * 16;
  size_t rowbase = ((size_t)g0 * L1 + j) * 64;
  const float* A1 = X0 + (size_t)g0 * 64;          // ld 64
  const float* A2 = Loc + rowbase;                 // ld L1*64
  const float* A3 = u + rowbase;                   // ld L1*64
  const float* B1 = W + (size_t)j * 64 * 64;       // rows (j*64+p), NxK
  v8f acc[4] = {};
#pragma unroll
  for (int k0 = 0; k0 < 64; k0 += 4) {
    v2f a1 = frag_mk(A1 + k0, 64, lane);
    v2f a2 = frag_mk(A2 + k0, L1 * 64, lane);
    v2f a3 = frag_mk(A3 + k0, L1 * 64, lane);
#pragma unroll
    for (int nt = 0; nt < 4; ++nt) {
      int p0 = nt * 16;
      v2f b1 = frag_mk(B1 + p0 * 64 + k0, 64, lane);
      v2f b2 = frag_mk(C + p0 * 64 + k0, 64, lane);
      v2f b3 = frag_mk(D + p0 * 64 + k0, 64, lane);
      acc[nt] = wmma4(a1, b1, acc[nt]);
      acc[nt] = wmma4(a2, b2, acc[nt]);
      acc[nt] = wmma4(a3, b3, acc[nt]);
    }
  }
#pragma unroll
  for (int nt = 0; nt < 4; ++nt)
    store_tile(out + rowbase + nt * 16, L1 * 64, acc[nt], lane);
}

// ---------------- host orchestration ------------------------------------------
static inline int imin_(int a, int b) { return a < b ? a : b; }

extern "C" void kernel_launch(void* const* d_in, const int* in_sizes, int n_in,
                              void* d_out, int out_size, void* d_ws, size_t ws_size,
                              hipStream_t stream) {
  (void)in_sizes; (void)n_in; (void)out_size; (void)ws_size;
  const float* u  = (const float*)d_in[0];
  const float* x0 = (const float*)d_in[1];
  const float* A  = (const float*)d_in[2];
  const float* B  = (const float*)d_in[3];
  const float* C  = (const float*)d_in[4];
  const float* D  = (const float*)d_in[5];
  float* out = (float*)d_out;

  // workspace layout (floats); total ~18.4M floats ~= 74 MB
  float* ws   = (float*)d_ws;
  float* Bu   = ws;                               // T*64       (drives -> local states)
  float* Pow  = Bu + (size_t)TLEN * 64;           // 129 * 4096 (A^0..A^128)
  float* Pow2 = Pow + 129 * 4096;                 // 65 * 4096  ((A^128)^0..^64)
  float* W    = Pow2 + 65 * 4096;                 // 8192*64    (C @ A^j stacked)
  float* E    = W + 8192 * 64;                    // G1*64      (chunk ends -> loc2)
  float* E2   = E + (size_t)G1 * 64;              // G2*64      (super ends -> z_s)
  float* X0v  = E2 + (size_t)G2 * 64;             // G1*64

  // A^j table by log-doubling: Pow[m+i] = Pow[m] @ Pow[i]
  k_init_pow<<<32, 256, 0, stream>>>(A, Pow);
  for (int m = 1; m <= 64; m *= 2) {
    int cnt = imin_(m, 128 - m);
    k_mm64<<<cnt, 128, 0, stream>>>(Pow + (size_t)m * 4096, Pow,
                                    Pow + (size_t)(m + 1) * 4096, 1);
  }
  // (A^128)^j table
  k_init_pow<<<32, 256, 0, stream>>>(Pow + (size_t)128 * 4096, Pow2);
  for (int m = 1; m <= 32; m *= 2) {
    int cnt = imin_(m, 64 - m);
    k_mm64<<<cnt, 128, 0, stream>>>(Pow2 + (size_t)m * 4096, Pow2,
                                    Pow2 + (size_t)(m + 1) * 4096, 1);
  }
  // W[j] = C @ A^j  (8192x64, L2-resident for the epilogue)
  k_mm64<<<128, 128, 0, stream>>>(C, Pow, W, 0);

  // Bu = u @ B^T
  k_bu<<<TLEN / 16 / 8, 256, 0, stream>>>(u, B, Bu);

  // level-1 scan: 2048 waves x 128 steps; in-place loc, ends -> E
  k_scan<<<G1 / 8, 256, 0, stream>>>(A, Bu, E, nullptr, L1, G1);
  // level-2 scan over chunk ends with A^128: 32 waves x 64 steps; ends -> E2
  k_scan<<<4, 256, 0, stream>>>(Pow + (size_t)128 * 4096, E, E2, nullptr, L2, G2);
  // level-3: single wave, seeded with x0; E2 becomes true super-chunk states z_s
  k_scan<<<1, 32, 0, stream>>>(Pow2 + (size_t)64 * 4096, E2, nullptr, x0, G2, 1);

  // true chunk-initial states: X0[g] = loc2[g] + (A^128)^(g%64) z_{g/64}
  k_x0<<<G1, 64, 0, stream>>>(Pow2, E2, E, X0v);

  // fused output: ys = Loc C^T + X0 (C A^j)^T + u D^T  (single write of d_out)
  k_out<<<(G1 / 16) * L1 / 8, 256, 0, stream>>>(u, Bu, X0v, W, C, D, out);
}